// MultiModel_12283606468029
// MI455X (gfx1250) — compile-verified
//
#include <hip/hip_runtime.h>

// ---------------------------------------------------------------------------
// MoE pricing kernel for MI455X (gfx1250, wave32).
//  - traffic (25MB) is L2-resident; kernel is transcendental-bound
//  - prep kernel hoists all parameter-only exp() and folds log2e/ln2/0.5
//  - hot loop uses native v_exp_f32 / v_log_f32 / v_tanh_f32 / v_rcp_f32
//  - gating matmul logits^T = W^T(9x5) @ h^T(5x16) via V_WMMA_F32_16X16X4_F32
//    (bias folded into the C operand); half-wave swaps via v_permlanex16
//  - 2 elements per thread per iteration to amortize constant readlanes
// ---------------------------------------------------------------------------

typedef __attribute__((ext_vector_type(2))) float v2f;
typedef __attribute__((ext_vector_type(8))) float v8f;

#define L2E 1.44269504088896340736f
#define LN2 0.69314718055994530942f

#if __has_builtin(__builtin_amdgcn_tanhf)
#define USE_TANH 1
#else
#define USE_TANH 0
#endif

#if __has_builtin(__builtin_amdgcn_exp2f)
#define EXP2F(x) __builtin_amdgcn_exp2f(x)
#else
#define EXP2F(x) __expf((x) * LN2)
#endif
#if __has_builtin(__builtin_amdgcn_logf)
#define LOG2F(x) __builtin_amdgcn_logf(x)
#else
#define LOG2F(x) (__logf(x) * L2E)
#endif
#if __has_builtin(__builtin_amdgcn_rcpf)
#define RCPF(x) __builtin_amdgcn_rcpf(x)
#else
#define RCPF(x) (1.0f / (x))
#endif

// Half-wave swap: lane L <-> lane L^16 (pure VALU on gfx12-family)
__device__ __forceinline__ float swap16(float v) {
#if __has_builtin(__builtin_amdgcn_permlanex16)
  int i = __float_as_int(v);
  int r = __builtin_amdgcn_permlanex16(i, i, 0x76543210, (int)0xfedcba98u,
                                       false, false);
  return __int_as_float(r);
#else
  return __shfl_xor(v, 16);
#endif
}

// Constant-buffer layout in d_ws (floats)
enum {
  WLOG = 0,    // 45: hid_out_weights[k*9+i] * log2e  (k=0..4)
  HOB  = 48,   // 9 : hid_out_bias * log2e
  GW0  = 57,   // 5 : in_hid_weights[k][0] * gs
  GW1  = 62,   // 5 : in_hid_weights[k][1] * gs
  GB   = 67,   // 5 : in_hid_bias * gs
  IB2  = 72,   // 45: invm_bias * log2e
  EWN  = 117,  // 45: -exp(invm_weights) * log2e
  TB2  = 162,  // 45: tau_bias scaled (see prep)
  ET2  = 207,  // 45: exp(tau_weights) scaled
  EP2  = 252,  // 45: exp(pricing_weights) * ln2 (* 0.5 for tanh path)
  NCONST = 297
};

__global__ void moe_prep(const float* __restrict__ ihw, const float* __restrict__ ihb,
                         const float* __restrict__ how, const float* __restrict__ hob,
                         const float* __restrict__ ib,  const float* __restrict__ iw,
                         const float* __restrict__ tb,  const float* __restrict__ tw,
                         const float* __restrict__ pw,  float* __restrict__ c) {
  int t = threadIdx.x;
  if (t < 45) {
    c[WLOG + t] = how[t] * L2E;
    c[IB2 + t]  = ib[t] * L2E;
    c[EWN + t]  = -__expf(iw[t]) * L2E;
#if USE_TANH
    c[TB2 + t]  = 0.5f * tb[t];
    c[ET2 + t]  = 0.5f * __expf(tw[t]);
    c[EP2 + t]  = 0.5f * LN2 * __expf(pw[t]);
#else
    c[TB2 + t]  = -tb[t] * L2E;
    c[ET2 + t]  = -__expf(tw[t]) * L2E;
    c[EP2 + t]  = LN2 * __expf(pw[t]);
#endif
  }
  if (t < 9) c[HOB + t] = hob[t] * L2E;
  if (t < 5) {
#if USE_TANH
    const float gs = 0.5f;
#else
    const float gs = -L2E;
#endif
    c[GW0 + t] = ihw[2 * t]     * gs;
    c[GW1 + t] = ihw[2 * t + 1] * gs;
    c[GB  + t] = ihb[t]         * gs;
  }
}

__global__ __launch_bounds__(256)
void moe_main(const float* __restrict__ invm, const float* __restrict__ tau,
              const float* __restrict__ cw, float* __restrict__ out, int iters) {
  const int lane = threadIdx.x & 31;
  const bool lo  = lane < 16;

  // A = W^T * log2e in the 16x4 f32 A-layout (loop invariant):
  // lane m (0..15): vgpr0=K0 row m, vgpr1=K1; lanes 16..31: K2, K3 of row m-16.
  const int row = lane & 15;
  const int kb  = (lane >> 4) * 2;  // 0 or 2
  v2f A;
  A.x = (row < 9) ? cw[WLOG + kb * 9 + row]       : 0.0f;
  A.y = (row < 9) ? cw[WLOG + (kb + 1) * 9 + row] : 0.0f;

  // C = bias (log2-scaled) broadcast in the C layout; rows 9..15 are junk and
  // never read, feed 0 there.
  v8f Cb;
#pragma unroll
  for (int r = 0; r < 8; ++r)
    Cb[r] = lo ? cw[HOB + r] : (r == 0 ? cw[HOB + 8] : 0.0f);

  int idx = blockIdx.x * blockDim.x + threadIdx.x;
  const int stride = gridDim.x * blockDim.x;

  for (int it = 0; it < iters; ++it, idx += 2 * stride) {
    float x[2], t[2];
    x[0] = invm[idx];          t[0] = tau[idx];
    x[1] = invm[idx + stride]; t[1] = tau[idx + stride];

    float acc[2], esum[2];

#pragma unroll
    for (int p = 0; p < 2; ++p) {
      // ---- gating hidden layer: h_k = sigmoid(w0*x + w1*t + b) ----
      float h[5];
#pragma unroll
      for (int k = 0; k < 5; ++k) {
        float z = fmaf(cw[GW0 + k], x[p], fmaf(cw[GW1 + k], t[p], cw[GB + k]));
#if USE_TANH
        h[k] = fmaf(0.5f, __builtin_amdgcn_tanhf(z), 0.5f);
#else
        h[k] = RCPF(1.0f + EXP2F(z));  // z = -z_raw*log2e
#endif
      }

      // ---- B tiles (4x16 = h^T): vgpr0 = K0/K2 rows, vgpr1 = K1/K3 ----
      const float m0 = swap16(h[0]);
      const float m1 = swap16(h[1]);
      const float m2 = swap16(h[2]);
      const float m3 = swap16(h[3]);
      v2f B0, B1;
      B0.x = lo ? h[0] : m2;  B0.y = lo ? h[1] : m3;
      B1.x = lo ? m0 : h[2];  B1.y = lo ? m1 : h[3];

      // D[i][n] = bias[i] + sum_{k<4} W[k][i]*h_k[n]   (all * log2e)
      v8f D0 = __builtin_amdgcn_wmma_f32_16x16x4_f32(false, A, false, B0,
                                                     (short)0, Cb, false, false);
      v8f D1 = __builtin_amdgcn_wmma_f32_16x16x4_f32(false, A, false, B1,
                                                     (short)0, Cb, false, false);

      // ---- gather logits per lane, add the k=4 term ----
      float l[9];
#pragma unroll
      for (int i = 0; i < 8; ++i) {
        float d1m = swap16(D1[i]);                 // lanes>=16 fetch their row
        float li  = lo ? D0[i] : d1m;
        l[i] = fmaf(h[4], cw[WLOG + 36 + i], li);
      }
      {
        float d0m = swap16(D0[0]);                 // M=8 lives in lanes 16-31
        float li  = lo ? d0m : D1[0];
        l[8] = fmaf(h[4], cw[WLOG + 36 + 8], li);
      }

      // ---- softmax numerators + expert mixture ----
      float es = 0.0f, ac = 0.0f;
#pragma unroll
      for (int i = 0; i < 9; ++i) {
        const float e = EXP2F(l[i]);
        es += e;
        float yi = 0.0f;
#pragma unroll
        for (int j = 0; j < 5; ++j) {
          const int c5 = i * 5 + j;
          // softplus(a)/ln2 = log2(1 + 2^(a*log2e))
          float u  = EXP2F(fmaf(x[p], cw[EWN + c5], cw[IB2 + c5]));
          float sp = LOG2F(1.0f + u);
          float b  = fmaf(t[p], cw[ET2 + c5], cw[TB2 + c5]);
#if USE_TANH
          float sg = 1.0f + __builtin_amdgcn_tanhf(b);   // 2*sigmoid; 0.5 in EP2
#else
          float sg = RCPF(1.0f + EXP2F(b));              // b = -b_raw*log2e
#endif
          yi = fmaf(sp * sg, cw[EP2 + c5], yi);          // EP2 carries ln2 (0.5)
        }
        ac = fmaf(yi, e, ac);
      }
      acc[p] = ac;
      esum[p] = es;
    }

    out[idx]          = acc[0] * RCPF(esum[0]);
    out[idx + stride] = acc[1] * RCPF(esum[1]);
  }
}

extern "C" void kernel_launch(void* const* d_in, const int* in_sizes, int n_in,
                              void* d_out, int out_size, void* d_ws, size_t ws_size,
                              hipStream_t stream) {
  const float* invm = (const float*)d_in[0];
  const float* tau  = (const float*)d_in[1];
  const float* ihw  = (const float*)d_in[2];
  const float* ihb  = (const float*)d_in[3];
  const float* how  = (const float*)d_in[4];
  const float* hob  = (const float*)d_in[5];
  const float* ib   = (const float*)d_in[6];
  const float* iw   = (const float*)d_in[7];
  const float* tb   = (const float*)d_in[8];
  const float* tw   = (const float*)d_in[9];
  const float* pw   = (const float*)d_in[10];
  float* out = (float*)d_out;
  float* cws = (float*)d_ws;

  const int n = in_sizes[0];  // 2^21: divides threads*blocks*2 exactly
  moe_prep<<<1, 64, 0, stream>>>(ihw, ihb, how, hob, ib, iw, tb, tw, pw, cws);

  const int threads = 256;
  int blocks = 1024;
  int iters = n / (threads * blocks * 2);
  if (iters < 1) iters = 1;
  moe_main<<<blocks, threads, 0, stream>>>(invm, tau, cws, out, iters);
}